// BasicBlock1D_27900107554824
// MI455X (gfx1250) — compile-verified
//
#include <hip/hip_runtime.h>

// ---------------------------------------------------------------------------
// BasicBlock1D for MI455X (gfx1250): per-position channel GEMMs on
// V_WMMA_F32_16X16X4_F32 with software-pipelined, double-buffered LDS
// (global prefetch of K-slice s+1 overlapped with WMMA compute of slice s),
// deterministic two-stage BN reductions, fused BN+ReLU into the second
// GEMM's A-operand load, LDS-tiled transposes for coalesced global traffic.
//
// Shapes (fixed by the reference): N=256, C=512, H=64.
// ---------------------------------------------------------------------------

typedef float v2f __attribute__((ext_vector_type(2)));
typedef float v8f __attribute__((ext_vector_type(8)));

#define N_ 256
#define C_ 512
#define H_ 64

// ---------------------------------------------------------------------------
// Generic (R x 64) -> (64 x R) transpose through LDS, 32x32 tiles.
//   xT[h][n][c]  = x[n][c][h]      (view x  as (N*C, H))
//   wT[h][o][c]  = w[o][c][h]      (view w  as (C*C, H))
// Both read and write sides are 128B-coalesced.
// ---------------------------------------------------------------------------
__global__ __launch_bounds__(256) void transpose_RxH(
    const float* __restrict__ in, float* __restrict__ out, int R) {
  __shared__ float tile[32][33];
  const int rb = blockIdx.x * 32;
  const int hb = blockIdx.y * 32;
  const int tx = threadIdx.x;      // 0..31
  const int ty = threadIdx.y;      // 0..7
  for (int j = 0; j < 32; j += 8)
    tile[ty + j][tx] = in[(size_t)(rb + ty + j) * H_ + hb + tx];
  __syncthreads();
  for (int j = 0; j < 32; j += 8)
    out[(size_t)(hb + ty + j) * R + rb + tx] = tile[tx][ty + j];
}

// ---------------------------------------------------------------------------
// Batched GEMM: for h in [0,64):  Y[h] (256x512) = A[h] (256x512) * B[h]
// with B[c][o] = W[h][o][c]  (W stored (H, O, C), K-contiguous rows).
// Block: 256 threads = 8 waves (2x4), wave tile 64x64, block tile 128x256.
// K pipelined in BK=16 slices through ping-pong LDS: while slice s is
// consumed by WMMAs, slice s+1 streams global->VGPR->LDS. One barrier per
// slice. Optional fused y = relu(a*scl[c]+bia[c]) on A-tile stores.
// ---------------------------------------------------------------------------
__global__ __launch_bounds__(256) void lc_gemm(
    const float* __restrict__ A,    // (H, N_, C_)
    const float* __restrict__ W,    // (H, C_, C_) : [h][o][c]
    float* __restrict__ Y,          // (H, N_, C_)
    const float* __restrict__ scl,  // per-K scale (fused BN) or nullptr
    const float* __restrict__ bia,  // per-K bias  (fused BN) or nullptr
    int fuse) {
  constexpr int BK = 16;
  constexpr int LDA = 20;  // LDS row stride: float4-aligned, conflict-free
  constexpr int ASZ = 128 * LDA;
  constexpr int BSZ = 256 * LDA;
  constexpr int S = C_ / BK;  // 32 K-slices
  __shared__ float sA[2 * ASZ];
  __shared__ float sB[2 * BSZ];

  const int h  = blockIdx.z;
  const int m0 = blockIdx.y * 128;
  const int n0 = blockIdx.x * 256;
  const int t    = threadIdx.x;
  const int lane = t & 31;
  const int wave = t >> 5;
  const int wm = (wave >> 2) * 64;  // wave M offset (2 waves)
  const int wn = (wave & 3) * 64;   // wave N offset (4 waves)

  const float* Ah = A + (size_t)h * N_ * C_;
  const float* Wh = W + (size_t)h * C_ * C_;

  v8f acc[4][4];
  const v8f vzero = {0.f, 0.f, 0.f, 0.f, 0.f, 0.f, 0.f, 0.f};
  for (int i = 0; i < 4; ++i)
    for (int j = 0; j < 4; ++j) acc[i][j] = vzero;

  const int la_r = t >> 2;        // 0..63: row within tile
  const int la_c = (t & 3) * 4;   // 0,4,8,12: col (float4) within K-slice

  float4 pa[2], pb[4];  // per-thread prefetch registers

  // ---- prologue: fetch slice 0 and stage it into buffer 0 ----
  for (int j = 0; j < 2; ++j)
    pa[j] = *(const float4*)(Ah + (size_t)(m0 + la_r + j * 64) * C_ + la_c);
  for (int j = 0; j < 4; ++j)
    pb[j] = *(const float4*)(Wh + (size_t)(n0 + la_r + j * 64) * C_ + la_c);
  {
    float4 va[2] = {pa[0], pa[1]};
    if (fuse) {
      const int kg = la_c;  // k0 == 0
      for (int j = 0; j < 2; ++j) {
        va[j].x = fmaxf(va[j].x * scl[kg + 0] + bia[kg + 0], 0.f);
        va[j].y = fmaxf(va[j].y * scl[kg + 1] + bia[kg + 1], 0.f);
        va[j].z = fmaxf(va[j].z * scl[kg + 2] + bia[kg + 2], 0.f);
        va[j].w = fmaxf(va[j].w * scl[kg + 3] + bia[kg + 3], 0.f);
      }
    }
    for (int j = 0; j < 2; ++j)
      *(float4*)(sA + (la_r + j * 64) * LDA + la_c) = va[j];
    for (int j = 0; j < 4; ++j)
      *(float4*)(sB + (la_r + j * 64) * LDA + la_c) = pb[j];
  }

  for (int s = 0; s < S; ++s) {
    const float* cA = sA + (s & 1) * ASZ;
    const float* cB = sB + (s & 1) * BSZ;
    __syncthreads();  // slice s staged; everyone done reading buffer (s+1)&1

    // ---- prefetch slice s+1 (overlaps with compute below) ----
    if (s + 1 < S) {
      const int kn = (s + 1) * BK;
      for (int j = 0; j < 2; ++j)
        pa[j] = *(const float4*)(Ah + (size_t)(m0 + la_r + j * 64) * C_ + kn +
                                 la_c);
      for (int j = 0; j < 4; ++j)
        pb[j] = *(const float4*)(Wh + (size_t)(n0 + la_r + j * 64) * C_ + kn +
                                 la_c);
    }

    // ---- compute slice s: 4 K=4 substeps ----
    for (int kk = 0; kk < BK; kk += 4) {
      // A 16x4 layout: lanes 0-15 hold K=kk..kk+1, lanes 16-31 K=kk+2..kk+3
      const int kf   = kk + ((lane >> 4) << 1);
      const int lrow = lane & 15;
      v2f af[4], bf[4];
      for (int mi = 0; mi < 4; ++mi)
        af[mi] = *(const v2f*)(cA + (wm + mi * 16 + lrow) * LDA + kf);
      for (int ni = 0; ni < 4; ++ni)
        bf[ni] = *(const v2f*)(cB + (wn + ni * 16 + lrow) * LDA + kf);
      for (int mi = 0; mi < 4; ++mi)
        for (int ni = 0; ni < 4; ++ni)
          acc[mi][ni] = __builtin_amdgcn_wmma_f32_16x16x4_f32(
              false, af[mi], false, bf[ni], (short)0, acc[mi][ni],
              false, false);
    }

    // ---- stage slice s+1 into the other buffer (guarded by next sync) ----
    if (s + 1 < S) {
      float* nA = sA + ((s + 1) & 1) * ASZ;
      float* nB = sB + ((s + 1) & 1) * BSZ;
      float4 va[2] = {pa[0], pa[1]};
      if (fuse) {
        const int kg = (s + 1) * BK + la_c;
        for (int j = 0; j < 2; ++j) {
          va[j].x = fmaxf(va[j].x * scl[kg + 0] + bia[kg + 0], 0.f);
          va[j].y = fmaxf(va[j].y * scl[kg + 1] + bia[kg + 1], 0.f);
          va[j].z = fmaxf(va[j].z * scl[kg + 2] + bia[kg + 2], 0.f);
          va[j].w = fmaxf(va[j].w * scl[kg + 3] + bia[kg + 3], 0.f);
        }
      }
      for (int j = 0; j < 2; ++j)
        *(float4*)(nA + (la_r + j * 64) * LDA + la_c) = va[j];
      for (int j = 0; j < 4; ++j)
        *(float4*)(nB + (la_r + j * 64) * LDA + la_c) = pb[j];
    }
  }

  // ---- store C tiles: VGPR r -> (M=r | r+8, N=lane%16) ----
  float* Yh = Y + (size_t)h * N_ * C_;
  const int rbase = (lane >> 4) * 8;
  const int ncol  = lane & 15;
  for (int mi = 0; mi < 4; ++mi)
    for (int ni = 0; ni < 4; ++ni) {
      const int gm = m0 + wm + mi * 16 + rbase;
      const int gn = n0 + wn + ni * 16 + ncol;
      const v8f c = acc[mi][ni];
      for (int r = 0; r < 8; ++r)
        Yh[(size_t)(gm + r) * C_ + gn] = c[r];
    }
}

// ---------------------------------------------------------------------------
// BN stats stage 1: per (h, n-chunk) partial sum / sumsq over each channel.
// Y layout (H, N, C): channel innermost -> coalesced. grid (H, 4), 256 thr.
// Deterministic fixed-order reduction (no float atomics).
// ---------------------------------------------------------------------------
__global__ __launch_bounds__(256) void stats_partial(
    const float* __restrict__ Y, float* __restrict__ psum,
    float* __restrict__ psq) {
  const int h = blockIdx.x;
  const int chunk = blockIdx.y;  // n in [chunk*64, chunk*64+64)
  const int t = threadIdx.x;
  for (int half = 0; half < 2; ++half) {
    const int c = t + half * 256;
    float s = 0.f, q = 0.f;
    const int nb = chunk * 64;
    for (int n = nb; n < nb + 64; ++n) {
      const float v = Y[(size_t)h * N_ * C_ + (size_t)n * C_ + c];
      s += v;
      q += v * v;
    }
    const int idx = (h * 4 + chunk) * C_ + c;
    psum[idx] = s;
    psq[idx]  = q;
  }
}

// ---------------------------------------------------------------------------
// BN stats stage 2: fold 256 partials per channel -> scale/bias.
// scale = g * rsqrt(var + eps); bias = b - mean*scale. One block, 512 thr.
// ---------------------------------------------------------------------------
__global__ __launch_bounds__(512) void stats_finalize(
    const float* __restrict__ psum, const float* __restrict__ psq,
    const float* __restrict__ g, const float* __restrict__ b,
    float* __restrict__ scale, float* __restrict__ bias) {
  const int c = threadIdx.x;
  float s = 0.f, q = 0.f;
  for (int i = 0; i < H_ * 4; ++i) {
    s += psum[i * C_ + c];
    q += psq[i * C_ + c];
  }
  const float inv = 1.f / (float)(N_ * H_);
  const float mean = s * inv;
  const float var  = q * inv - mean * mean;
  const float sc   = g[c] * rsqrtf(var + 1e-5f);
  scale[c] = sc;
  bias[c]  = b[c] - mean * sc;
}

// ---------------------------------------------------------------------------
// Final: out[n][c][h] = relu(Y2T[h][n][c]*scale[c] + bias[c] + x[n][c][h]).
// (H,N,C) -> (N,C,H) transpose via 64x64 LDS tile per n; both the Y2 read
// (contiguous in c) and the x/out access (contiguous in h) are coalesced.
// grid = N * (C/64), 256 threads.
// ---------------------------------------------------------------------------
__global__ __launch_bounds__(256) void bn_residual_relu(
    const float* __restrict__ Y2T, const float* __restrict__ x,
    const float* __restrict__ scale, const float* __restrict__ bias,
    float* __restrict__ out) {
  __shared__ float tile[64 * 65];
  const int n  = blockIdx.x >> 3;
  const int c0 = (blockIdx.x & 7) * 64;
  const int t  = threadIdx.x;

  const int cl = t & 63, h0 = t >> 6;  // load: c fastest
  for (int j = 0; j < 64; j += 4) {
    const int h = h0 + j;
    tile[cl * 65 + h] = Y2T[(size_t)h * N_ * C_ + (size_t)n * C_ + c0 + cl];
  }
  __syncthreads();

  const int h = t & 63, cl2 = t >> 6;  // store: h fastest
  for (int j = 0; j < 64; j += 4) {
    const int c = c0 + cl2 + j;
    const float v = tile[(cl2 + j) * 65 + h];
    const size_t idx = ((size_t)n * C_ + c) * H_ + h;
    const float r = v * scale[c] + bias[c] + x[idx];
    out[idx] = fmaxf(r, 0.f);
  }
}

// ---------------------------------------------------------------------------
// Host-side pipeline (all on `stream`, graph-capture safe).
// Workspace layout (floats):
//   bufA  [0,          8388608)   xT, later reused as Y2T
//   bufW  [8388608,   25165824)   w1T, later reused for w2T
//   bufY1 [25165824,  33554432)   Y1T
//   psum  [33554432,  33685504)
//   psq   [33685504,  33816576)
//   scale1/bias1/scale2/bias2: 4*512 floats after that.   Total ~135.3 MB.
// ---------------------------------------------------------------------------
extern "C" void kernel_launch(void* const* d_in, const int* in_sizes, int n_in,
                              void* d_out, int out_size, void* d_ws,
                              size_t ws_size, hipStream_t stream) {
  const float* x  = (const float*)d_in[0];
  const float* w1 = (const float*)d_in[1];
  const float* g1 = (const float*)d_in[2];
  const float* b1 = (const float*)d_in[3];
  const float* w2 = (const float*)d_in[4];
  const float* g2 = (const float*)d_in[5];
  const float* b2 = (const float*)d_in[6];
  float* out = (float*)d_out;

  float* ws     = (float*)d_ws;
  float* bufA   = ws;                       // 8,388,608
  float* bufW   = ws + 8388608;             // 16,777,216
  float* bufY1  = ws + 25165824;            // 8,388,608
  float* psum   = ws + 33554432;            // 131,072
  float* psq    = ws + 33685504;            // 131,072
  float* scale1 = ws + 33816576;
  float* bias1  = scale1 + C_;
  float* scale2 = bias1 + C_;
  float* bias2  = scale2 + C_;

  const dim3 tb(32, 8);

  // x -> xT (H, N, C)
  transpose_RxH<<<dim3(N_ * C_ / 32, H_ / 32), tb, 0, stream>>>(x, bufA,
                                                                N_ * C_);
  // w1 -> w1T (H, O, C)
  transpose_RxH<<<dim3(C_ * C_ / 32, H_ / 32), tb, 0, stream>>>(w1, bufW,
                                                                C_ * C_);
  // conv1: Y1T = xT @ w1T
  lc_gemm<<<dim3(2, 2, H_), 256, 0, stream>>>(bufA, bufW, bufY1, nullptr,
                                              nullptr, 0);
  // BN1 stats
  stats_partial<<<dim3(H_, 4), 256, 0, stream>>>(bufY1, psum, psq);
  stats_finalize<<<1, 512, 0, stream>>>(psum, psq, g1, b1, scale1, bias1);
  // w2 -> w2T (reuse bufW)
  transpose_RxH<<<dim3(C_ * C_ / 32, H_ / 32), tb, 0, stream>>>(w2, bufW,
                                                                C_ * C_);
  // conv2 with fused BN1+ReLU on A; Y2T reuses bufA (xT fully consumed)
  lc_gemm<<<dim3(2, 2, H_), 256, 0, stream>>>(bufY1, bufW, bufA, scale1,
                                              bias1, 1);
  // BN2 stats
  stats_partial<<<dim3(H_, 4), 256, 0, stream>>>(bufA, psum, psq);
  stats_finalize<<<1, 512, 0, stream>>>(psum, psq, g2, b2, scale2, bias2);
  // BN2 + residual + ReLU, (H,N,C) -> (N,C,H)
  bn_residual_relu<<<dim3(N_ * (C_ / 64)), 256, 0, stream>>>(bufA, x, scale2,
                                                             bias2, out);
}